// TopKRouter_66365834657998
// MI455X (gfx1250) — compile-verified
//
#include <hip/hip_runtime.h>
#include <hip/hip_bf16.h>

// ---------------------------------------------------------------------------
// MoE Top-K router for MI455X (gfx1250, wave32).
// logits = x @ W^T + b  (fp32 WMMA 16x16x4), softmax, top-2, capacity scatter.
// Double-buffered K-chunks staged via async global->LDS copies (ASYNCcnt).
// ---------------------------------------------------------------------------

typedef __attribute__((ext_vector_type(2))) float v2f;
typedef __attribute__((ext_vector_type(8))) float v8f;
typedef __attribute__((ext_vector_type(4))) int   v4i;

typedef __attribute__((address_space(1))) v4i glb_v4i;   // global int4
typedef __attribute__((address_space(3))) v4i lds_v4i;   // LDS int4

#define NEXP   64      // experts (fixed by problem)
#define KTOP   2       // top-k
#define CAP    640     // expert capacity
#define KC     64      // K-chunk staged through LDS
#define LDP    68      // padded LDS row stride (floats); 68*4=272 B, 16B-aligned
#define SCAP   4096    // max assignments per expert held in LDS (avg ~512)

// ---- async global->LDS copy (CDNA5) with safe fallback --------------------
#if __has_builtin(__builtin_amdgcn_global_load_async_to_lds_b128)
#define HAVE_ASYNC 1
#define ASYNC_CP_B128(gp, sp)                                                  \
    __builtin_amdgcn_global_load_async_to_lds_b128(                            \
        (glb_v4i*)(gp), (lds_v4i*)(sp), 0, 0)
#else
#define HAVE_ASYNC 0
#define ASYNC_CP_B128(gp, sp) (*(float4*)(sp) = *(const float4*)(gp))
#endif

#if __has_builtin(__builtin_amdgcn_s_wait_asynccnt)
#define WAIT_ASYNC(n) __builtin_amdgcn_s_wait_asynccnt(n)
#else
#define WAIT_ASYNC(n) asm volatile("s_wait_asynccnt %0" :: "i"(n) : "memory")
#endif

// ---------------------------------------------------------------------------
// Kernel 1: GEMM (WMMA f32) + bias + softmax + top-2 per row.
// Block: 256 threads = 8 waves. Block covers 32 rows x 64 experts.
// wave w: rowTile = w>>2 (16 rows), expertTile = w&3 (16 experts).
// ---------------------------------------------------------------------------
__global__ __launch_bounds__(256)
void router_gemm_softmax_k(const float* __restrict__ x,
                           const float* __restrict__ W,
                           const float* __restrict__ bias,
                           float* __restrict__ logits,
                           float* __restrict__ probs,
                           int*   __restrict__ a_eid,
                           float* __restrict__ a_un,
                           float* __restrict__ a_pn,
                           int D)
{
    __shared__ float xs[2][32 * LDP];   // x chunk: 32 rows x KC, double buffered
    __shared__ float ws[2][64 * LDP];   // W chunk: 64 experts x KC, double buffered
    __shared__ float ls[32 * LDP];      // biased logits staging: 32 rows x 64

    const int t    = threadIdx.x;
    const int lane = t & 31;
    const int wave = t >> 5;
    const int rt   = wave >> 2;      // row tile: 0..1
    const int et   = wave & 3;       // expert tile: 0..3
    const int r0   = blockIdx.x * 32;

    // fragment addressing (ISA 16x4 A layout / 4x16 B layout):
    // lanes 0-15 hold K={0,1}, lanes 16-31 hold K={2,3} of each 4-wide K step
    const int lrow  = lane & 15;
    const int khalf = (lane >> 4) * 2;

    // per-thread staging assignments (coalesced b128)
    const int xrow_g = t >> 3;            // 0..31
    const int xcol_g = (t & 7) * 8;       // 8 floats (2 x b128) per thread
    const int wrow_g = t >> 2;            // 0..63
    const int wcol_g = (t & 3) * 16;      // 16 floats (4 x b128) per thread

    const float* gx = x + (size_t)(r0 + xrow_g) * D + xcol_g;
    const float* gw = W + (size_t)wrow_g * D + wcol_g;
    float* sx0 = &xs[0][xrow_g * LDP + xcol_g];
    float* sx1 = &xs[1][xrow_g * LDP + xcol_g];
    float* sw0 = &ws[0][wrow_g * LDP + wcol_g];
    float* sw1 = &ws[1][wrow_g * LDP + wcol_g];

    v8f c = {};                            // 16x16 fp32 accumulator (8 VGPRs)

    // ---- prologue: stage chunk 0 into buffer 0 (6 async b128 per thread) ----
    {
        ASYNC_CP_B128(gx,     sx0);
        ASYNC_CP_B128(gx + 4, sx0 + 4);
#pragma unroll
        for (int q = 0; q < 4; ++q)
            ASYNC_CP_B128(gw + 4 * q, sw0 + 4 * q);
    }

    int buf = 0;
    for (int k0 = 0; k0 < D; k0 += KC) {
        const bool has_next = (k0 + KC) < D;
        if (has_next) {                    // stage next chunk into other buffer
            const float* gxn = gx + k0 + KC;
            const float* gwn = gw + k0 + KC;
            float* sxn = buf ? sx0 : sx1;
            float* swn = buf ? sw0 : sw1;
            ASYNC_CP_B128(gxn,     sxn);
            ASYNC_CP_B128(gxn + 4, sxn + 4);
#pragma unroll
            for (int q = 0; q < 4; ++q)
                ASYNC_CP_B128(gwn + 4 * q, swn + 4 * q);
            WAIT_ASYNC(6);                 // current stage (first 6, in order) done
        } else {
            WAIT_ASYNC(0);
        }
        __syncthreads();                   // all waves' LDS writes visible

        // ---- WMMA over the chunk: 16 x v_wmma_f32_16x16x4_f32 ----
        const float* xr = &xs[buf][(rt * 16 + lrow) * LDP];
        const float* wr = &ws[buf][(et * 16 + lrow) * LDP];
#pragma unroll
        for (int k = 0; k < KC; k += 4) {
            v2f a = *(const v2f*)(xr + k + khalf);   // A: x rows  (ds_load_b64)
            v2f b = *(const v2f*)(wr + k + khalf);   // B: W^T cols
            c = __builtin_amdgcn_wmma_f32_16x16x4_f32(
                    false, a, false, b, (short)0, c, false, false);
        }
        __syncthreads();                   // compute done before buf is re-staged
        buf ^= 1;
    }

    // ---- bias + write logits + stage for softmax ----
    const int ecol = et * 16 + lrow;
    const float bv = bias[ecol];
#pragma unroll
    for (int i = 0; i < 8; ++i) {
        // C layout: VGPR i holds M=i (lanes 0-15) and M=i+8 (lanes 16-31)
        const int row = rt * 16 + i + ((lane >> 4) << 3);
        const float v = c[i] + bv;
        logits[(size_t)(r0 + row) * NEXP + ecol] = v;
        ls[row * LDP + ecol] = v;
    }
    __syncthreads();

    // ---- softmax + top-2 (one thread per row; 32 rows) ----
    if (t < 32) {
        const float* lr = &ls[t * LDP];
        float m = lr[0];
#pragma unroll 8
        for (int e = 1; e < NEXP; ++e) m = fmaxf(m, lr[e]);
        float s = 0.f;
#pragma unroll 8
        for (int e = 0; e < NEXP; ++e) s += __expf(lr[e] - m);
        const float inv = 1.f / s;

        float p1 = -1.f, p2 = -1.f;
        int   e1 = -1,  e2 = -1;
        float* pout = probs + (size_t)(r0 + t) * NEXP;
#pragma unroll 8
        for (int e = 0; e < NEXP; ++e) {
            const float p = __expf(lr[e] - m) * inv;
            pout[e] = p;
            if (p > p1)      { p2 = p1; e2 = e1; p1 = p; e1 = e; }
            else if (p > p2) { p2 = p;  e2 = e; }
        }
        const float dnm = p1 + p2;
        const int fi = (r0 + t) * KTOP;
        a_eid[fi]     = e1; a_un[fi]     = p1; a_pn[fi]     = p1 / dnm;
        a_eid[fi + 1] = e2; a_un[fi + 1] = p2; a_pn[fi + 1] = p2 / dnm;
    }
}

// ---------------------------------------------------------------------------
// Kernel 2: per-expert capacity scatter. One block per expert.
// Rank(i) = #{j of same expert : p_j > p_i || (p_j == p_i && idx_j < idx_i)}
// == position after stable lexsort((-un, eid)) within the expert group.
// ---------------------------------------------------------------------------
__global__ __launch_bounds__(256)
void expert_scatter_k(const int*   __restrict__ a_eid,
                      const float* __restrict__ a_un,
                      const float* __restrict__ a_pn,
                      float* __restrict__ expert_probs,
                      int*   __restrict__ expert_indices,
                      int total)   // N * KTOP
{
    __shared__ float sp[SCAP];
    __shared__ float spn[SCAP];
    __shared__ int   sfi[SCAP];
    __shared__ int   cnt;

    const int e = blockIdx.x;
    const int t = threadIdx.x;
    if (t == 0) cnt = 0;
    __syncthreads();

    // compact this expert's assignments into LDS (order irrelevant)
    for (int i = t; i < total; i += 256) {
        if (a_eid[i] == e) {
            const int pos = atomicAdd(&cnt, 1);
            if (pos < SCAP) {
                sp[pos]  = a_un[i];
                spn[pos] = a_pn[i];
                sfi[pos] = i;
            }
        }
    }

    // initialize output row: probs=0, indices=-1
    for (int cc = t; cc < CAP; cc += 256) {
        expert_probs[(size_t)e * CAP + cc]   = 0.f;
        expert_indices[(size_t)e * CAP + cc] = -1;
    }
    __syncthreads();

    int M = cnt;
    if (M > SCAP) M = SCAP;

    for (int i = t; i < M; i += 256) {
        const float pi = sp[i];
        const int   fi = sfi[i];
        int rank = 0;
        for (int j = 0; j < M; ++j) {
            const float pj = sp[j];
            rank += (pj > pi) | ((pj == pi) & (sfi[j] < fi));
        }
        if (rank < CAP) {
            expert_probs[(size_t)e * CAP + rank]   = spn[i];
            expert_indices[(size_t)e * CAP + rank] = fi >> 1;  // token = flat / k
        }
    }
}

// ---------------------------------------------------------------------------
extern "C" void kernel_launch(void* const* d_in, const int* in_sizes, int n_in,
                              void* d_out, int out_size, void* d_ws, size_t ws_size,
                              hipStream_t stream)
{
    const float* x = (const float*)d_in[0];
    const float* W = (const float*)d_in[1];
    const float* b = (const float*)d_in[2];
    // d_in[3] = k (==2), d_in[4] = expert_capacity (==640): fixed by setup.

    const int E = in_sizes[2];                 // 64
    const int D = in_sizes[1] / E;             // 4096
    const int N = in_sizes[0] / D;             // 16384

    float* out    = (float*)d_out;
    float* logits = out;                                   // N*E
    float* probs  = out + (size_t)N * NEXP;                // N*E
    float* eprob  = probs + (size_t)N * NEXP;              // E*CAP
    int*   eidx   = (int*)(eprob + (size_t)NEXP * CAP);    // E*CAP (int32 bits)

    const int total = N * KTOP;
    int*   a_eid = (int*)d_ws;
    float* a_un  = (float*)(a_eid + total);
    float* a_pn  = a_un + total;

    router_gemm_softmax_k<<<N / 32, 256, 0, stream>>>(
        x, W, b, logits, probs, a_eid, a_un, a_pn, D);

    expert_scatter_k<<<NEXP, 256, 0, stream>>>(
        a_eid, a_un, a_pn, eprob, eidx, total);
}